// NodeModel_5042291605552
// MI455X (gfx1250) — compile-verified
//
#include <hip/hip_runtime.h>
#include <stdint.h>

#define C       128
#define NNODES  50000
#define NEDGES  800000
#define BN_EPS  1e-5f

// LDS row strides (floats): 136 = 128 data + 8 pad -> avoids 64-bank conflicts
#define AST 136
#define WST 136

typedef float v2f  __attribute__((ext_vector_type(2)));
typedef float v8f  __attribute__((ext_vector_type(8)));
typedef unsigned int u32x4 __attribute__((ext_vector_type(4)));
typedef int   i32x4 __attribute__((ext_vector_type(4)));
typedef int   i32x8 __attribute__((ext_vector_type(8)));

__device__ __forceinline__ v8f wmma4(v2f a, v2f b, v8f c) {
  // D = A(16x4 f32) * B(4x16 f32) + C(16x16 f32)
  return __builtin_amdgcn_wmma_f32_16x16x4_f32(false, a, false, b, (short)0, c,
                                               false, false);
}

__device__ __forceinline__ uint32_t lds_off(const void* p) {
  // low 32 bits of a flat shared-aperture address == LDS byte offset
  return (uint32_t)(uintptr_t)p;
}

// 2D tile load via Tensor Data Mover: tile_rows x tile_cols f32 elements,
// rows placed into LDS at (tile_cols + pad) dword stride via pad_enable.
__device__ __forceinline__ void tdm_load_2d(uint32_t lds_byte_off,
                                            const void* gptr,
                                            uint32_t tile_cols, uint32_t tile_rows,
                                            uint32_t tensor_cols, uint32_t tensor_rows,
                                            uint32_t row_stride_elems,
                                            uint32_t pad_interval_code,
                                            uint32_t pad_amount_code) {
  uint64_t ga = (uint64_t)(uintptr_t)gptr;
  u32x4 g0;
  g0[0] = 1u;                                     // count=1, user descriptor
  g0[1] = lds_byte_off;                           // lds_addr
  g0[2] = (uint32_t)ga;                           // global_addr lo
  g0[3] = (uint32_t)(ga >> 32) | (2u << 30);      // global_addr hi | type=2
  i32x8 g1;
  g1[0] = (int)((2u << 16) |                      // data_size = 4B
                (1u << 20) |                      // pad_enable
                (pad_interval_code << 22) |
                (pad_amount_code << 25));
  g1[1] = (int)((tensor_cols & 0xFFFFu) << 16);                               // dim0 lo
  g1[2] = (int)(((tensor_cols >> 16) & 0xFFFFu) | ((tensor_rows & 0xFFFFu) << 16)); // dim0 hi | dim1 lo
  g1[3] = (int)(((tensor_rows >> 16) & 0xFFFFu) | ((tile_cols & 0xFFFFu) << 16));   // dim1 hi | tile_dim0
  g1[4] = (int)(tile_rows & 0xFFFFu);             // tile_dim1 (tile_dim2 = 0)
  g1[5] = (int)row_stride_elems;                  // tensor_dim0_stride lo
  g1[6] = 0;                                      // stride hi | dim1_stride lo
  g1[7] = 0;
  i32x4 z4 = {0, 0, 0, 0};
  i32x8 z8 = {0, 0, 0, 0, 0, 0, 0, 0};            // groups 2+3 unused (2D tensor)
  __builtin_amdgcn_tensor_load_to_lds(g0, g1, z4, z4, z8, 0);
}

__device__ __forceinline__ float elu(float v) {
  return v > 0.0f ? v : (__expf(v) - 1.0f);
}

// ---------------------------------------------------------------------------
// Edge kernels: h = ELU(BN(concat(x[row], edge_attr) @ w1.T + b1))
// PASS 0: accumulate per-channel sum / sumsq of pre-BN h into stats[0..2C)
// PASS 1: recompute, apply BN+ELU, scatter-add into agg / cnt
// ---------------------------------------------------------------------------
template <int PASS>
__global__ __launch_bounds__(256) void edge_kernel(
    const float* __restrict__ x, const int* __restrict__ eidx,
    const float* __restrict__ eattr, const float* __restrict__ w1,
    const float* __restrict__ b1, const float* __restrict__ bn1w,
    const float* __restrict__ bn1b, float* __restrict__ stats,
    float* __restrict__ agg, float* __restrict__ cnt) {
  __shared__ float sAx[64 * AST];   // x[row] tile   (K = 0..127)
  __shared__ float sAe[64 * AST];   // edge_attr tile (K = 128..255)
  __shared__ float sW[128 * WST];   // half of w1 (128 rows x 128 cols)
  __shared__ int   sCol[64];

  const int tid  = threadIdx.x;
  const int lane = tid & 31;
  const int w    = tid >> 5;
  const int e0   = blockIdx.x * 64;

  if (w == 0) {
    // w1 cols [0,128): rows at LDS stride 136 (pad interval 128 dw, pad 8 dw)
    tdm_load_2d(lds_off(sW), w1, 128, 128, 256, 128, 256, 6, 7);
    // 64 contiguous edge_attr rows
    tdm_load_2d(lds_off(sAe), eattr + (size_t)e0 * C, 128, 64, 128,
                (uint32_t)(NEDGES - e0), 128, 6, 7);
  }

  {  // gather x[row] rows: 4 threads per row, 32 floats each
    int r = tid >> 2;
    int cb = (tid & 3) * 32;
    int row = eidx[e0 + r];
    const float4* src = (const float4*)(x + (size_t)row * C + cb);
    float* dst = sAx + r * AST + cb;
#pragma unroll
    for (int j = 0; j < 8; ++j) *(float4*)(dst + 4 * j) = src[j];
    if (tid < 64) sCol[tid] = eidx[NEDGES + e0 + tid];
  }

  if (w == 0) __builtin_amdgcn_s_wait_tensorcnt(0);
  __syncthreads();

  const int mt  = w & 3;            // 16-edge row tile
  const int ntb = (w >> 2) * 4;     // first of 4 16-channel col tiles
  const int lm  = lane & 15;
  const int lh  = lane >> 4;

  v8f acc[4];
#pragma unroll
  for (int j = 0; j < 4; ++j) acc[j] = (v8f){0, 0, 0, 0, 0, 0, 0, 0};

  // Phase 0: K = 0..127 (x part)
  const float* Arow = sAx + (mt * 16 + lm) * AST;
#pragma unroll 8
  for (int k0 = 0; k0 < 128; k0 += 4) {
    v2f a = *(const v2f*)(Arow + k0 + lh * 2);
#pragma unroll
    for (int j = 0; j < 4; ++j) {
      v2f b = *(const v2f*)(sW + ((ntb + j) * 16 + lm) * WST + k0 + lh * 2);
      acc[j] = wmma4(a, b, acc[j]);
    }
  }

  __syncthreads();
  if (w == 0) {  // restage w1 cols [128,256)
    tdm_load_2d(lds_off(sW), w1 + 128, 128, 128, 128, 128, 256, 6, 7);
    __builtin_amdgcn_s_wait_tensorcnt(0);
  }
  __syncthreads();

  // Phase 1: K = 128..255 (edge_attr part)
  const float* Arow2 = sAe + (mt * 16 + lm) * AST;
#pragma unroll 8
  for (int k0 = 0; k0 < 128; k0 += 4) {
    v2f a = *(const v2f*)(Arow2 + k0 + lh * 2);
#pragma unroll
    for (int j = 0; j < 4; ++j) {
      v2f b = *(const v2f*)(sW + ((ntb + j) * 16 + lm) * WST + k0 + lh * 2);
      acc[j] = wmma4(a, b, acc[j]);
    }
  }

  if (PASS == 0) {
#pragma unroll
    for (int j = 0; j < 4; ++j) {
      int n = (ntb + j) * 16 + lm;
      float bias = b1[n];
      float s = 0.0f, q = 0.0f;
#pragma unroll
      for (int i = 0; i < 8; ++i) {
        float v = acc[j][i] + bias;
        s += v;
        q += v * v;
      }
      s += __shfl_xor(s, 16, 32);
      q += __shfl_xor(q, 16, 32);
      if (lh == 0) {
        atomicAdd(&stats[n], s);
        atomicAdd(&stats[C + n], q);
      }
    }
  } else {
    const float invE = 1.0f / (float)NEDGES;
#pragma unroll
    for (int j = 0; j < 4; ++j) {
      int n = (ntb + j) * 16 + lm;
      float bias  = b1[n];
      float mean  = stats[n] * invE;
      float var   = stats[C + n] * invE - mean * mean;
      float scale = bn1w[n] * rsqrtf(var + BN_EPS);
      float shift = bn1b[n] - mean * scale;
#pragma unroll
      for (int i = 0; i < 8; ++i) {
        int m = mt * 16 + i + 8 * lh;  // edge within tile (C-layout: M = i + 8*half)
        float v = elu((acc[j][i] + bias) * scale + shift);
        atomicAdd(&agg[(size_t)sCol[m] * C + n], v);
      }
    }
    if (tid < 64) atomicAdd(&cnt[sCol[tid]], 1.0f);
  }
}

// ---------------------------------------------------------------------------
// Node kernel: pre2 = concat(x, agg/max(cnt,1)) @ w2.T + b2 ; stats2 += sums
// ---------------------------------------------------------------------------
__global__ __launch_bounds__(256) void node_kernel(
    const float* __restrict__ x, const float* __restrict__ agg,
    const float* __restrict__ cnt, const float* __restrict__ w2,
    const float* __restrict__ b2, float* __restrict__ stats2,
    float* __restrict__ pre2) {
  __shared__ float sAx[64 * AST];
  __shared__ float sAe[64 * AST];
  __shared__ float sW[128 * WST];

  const int tid  = threadIdx.x;
  const int lane = tid & 31;
  const int w    = tid >> 5;
  const int i0   = blockIdx.x * 64;

  if (w == 0) {
    tdm_load_2d(lds_off(sW), w2, 128, 128, 256, 128, 256, 6, 7);
    // contiguous x rows; OOB rows (beyond N) zero-filled by TDM range check
    tdm_load_2d(lds_off(sAx), x + (size_t)i0 * C, 128, 64, 128,
                (uint32_t)(NNODES - i0), 128, 6, 7);
  }

  {  // agg rows scaled by 1/max(cnt,1)
    int r = tid >> 2;
    int cb = (tid & 3) * 32;
    int row = i0 + r;
    int rc = row < NNODES ? row : NNODES - 1;
    float inv = 1.0f / fmaxf(cnt[rc], 1.0f);
    const float4* src = (const float4*)(agg + (size_t)rc * C + cb);
    float* dst = sAe + r * AST + cb;
#pragma unroll
    for (int j = 0; j < 8; ++j) {
      float4 v = src[j];
      v.x *= inv; v.y *= inv; v.z *= inv; v.w *= inv;
      *(float4*)(dst + 4 * j) = v;
    }
  }

  if (w == 0) __builtin_amdgcn_s_wait_tensorcnt(0);
  __syncthreads();

  const int mt  = w & 3;
  const int ntb = (w >> 2) * 4;
  const int lm  = lane & 15;
  const int lh  = lane >> 4;

  v8f acc[4];
#pragma unroll
  for (int j = 0; j < 4; ++j) acc[j] = (v8f){0, 0, 0, 0, 0, 0, 0, 0};

  const float* Arow = sAx + (mt * 16 + lm) * AST;
#pragma unroll 8
  for (int k0 = 0; k0 < 128; k0 += 4) {
    v2f a = *(const v2f*)(Arow + k0 + lh * 2);
#pragma unroll
    for (int j = 0; j < 4; ++j) {
      v2f b = *(const v2f*)(sW + ((ntb + j) * 16 + lm) * WST + k0 + lh * 2);
      acc[j] = wmma4(a, b, acc[j]);
    }
  }

  __syncthreads();
  if (w == 0) {
    tdm_load_2d(lds_off(sW), w2 + 128, 128, 128, 128, 128, 256, 6, 7);
    __builtin_amdgcn_s_wait_tensorcnt(0);
  }
  __syncthreads();

  const float* Arow2 = sAe + (mt * 16 + lm) * AST;
#pragma unroll 8
  for (int k0 = 0; k0 < 128; k0 += 4) {
    v2f a = *(const v2f*)(Arow2 + k0 + lh * 2);
#pragma unroll
    for (int j = 0; j < 4; ++j) {
      v2f b = *(const v2f*)(sW + ((ntb + j) * 16 + lm) * WST + k0 + lh * 2);
      acc[j] = wmma4(a, b, acc[j]);
    }
  }

#pragma unroll
  for (int j = 0; j < 4; ++j) {
    int n = (ntb + j) * 16 + lm;
    float bias = b2[n];
    float s = 0.0f, q = 0.0f;
#pragma unroll
    for (int i = 0; i < 8; ++i) {
      int row = i0 + mt * 16 + i + 8 * lh;
      float v = acc[j][i] + bias;
      if (row < NNODES) {
        pre2[(size_t)row * C + n] = v;
        s += v;
        q += v * v;
      }
    }
    s += __shfl_xor(s, 16, 32);
    q += __shfl_xor(q, 16, 32);
    if (lh == 0) {
      atomicAdd(&stats2[n], s);
      atomicAdd(&stats2[C + n], q);
    }
  }
}

// out = ELU(BN(pre2))
__global__ __launch_bounds__(256) void finalize_kernel(
    const float* __restrict__ pre2, const float* __restrict__ stats2,
    const float* __restrict__ bn2w, const float* __restrict__ bn2b,
    float* __restrict__ out, int total) {
  int idx = blockIdx.x * blockDim.x + threadIdx.x;
  if (idx >= total) return;
  int c = idx & (C - 1);
  const float invN = 1.0f / (float)NNODES;
  float mean  = stats2[c] * invN;
  float var   = stats2[C + c] * invN - mean * mean;
  float scale = bn2w[c] * rsqrtf(var + BN_EPS);
  float shift = bn2b[c] - mean * scale;
  out[idx] = elu(pre2[idx] * scale + shift);
}

extern "C" void kernel_launch(void* const* d_in, const int* in_sizes, int n_in,
                              void* d_out, int out_size, void* d_ws,
                              size_t ws_size, hipStream_t stream) {
  (void)in_sizes; (void)n_in; (void)out_size; (void)ws_size;
  const float* x     = (const float*)d_in[0];
  const int*   eidx  = (const int*)d_in[1];
  const float* eattr = (const float*)d_in[2];
  // d_in[3] (u) and d_in[4] (batch) are unused by the reference
  const float* w1    = (const float*)d_in[5];
  const float* b1    = (const float*)d_in[6];
  const float* bn1w  = (const float*)d_in[7];
  const float* bn1b  = (const float*)d_in[8];
  const float* w2    = (const float*)d_in[9];
  const float* b2    = (const float*)d_in[10];
  const float* bn2w  = (const float*)d_in[11];
  const float* bn2b  = (const float*)d_in[12];

  float* ws   = (float*)d_ws;
  float* agg  = ws;                        // N*C scatter sums
  float* cnt  = agg + (size_t)NNODES * C;  // N
  float* st1  = cnt + NNODES;              // 2C edge BN stats
  float* st2  = st1 + 2 * C;               // 2C node BN stats
  float* pre2 = st2 + 2 * C;               // N*C pre-BN node output

  size_t zero_floats = (size_t)NNODES * C + NNODES + 4 * C;
  (void)hipMemsetAsync(ws, 0, zero_floats * sizeof(float), stream);

  edge_kernel<0><<<NEDGES / 64, 256, 0, stream>>>(x, eidx, eattr, w1, b1, bn1w,
                                                  bn1b, st1, agg, cnt);
  edge_kernel<1><<<NEDGES / 64, 256, 0, stream>>>(x, eidx, eattr, w1, b1, bn1w,
                                                  bn1b, st1, agg, cnt);
  node_kernel<<<(NNODES + 63) / 64, 256, 0, stream>>>(x, agg, cnt, w2, b2, st2,
                                                      pre2);
  int total = NNODES * C;
  finalize_kernel<<<(total + 255) / 256, 256, 0, stream>>>(pre2, st2, bn2w,
                                                           bn2b, (float*)d_out,
                                                           total);
}